// HierarchicalAutoEncoder_90125593739622
// MI455X (gfx1250) — compile-verified
//
#include <hip/hip_runtime.h>

// HierarchicalAutoEncoder on MI455X (gfx1250), fp32 via V_WMMA_F32_16X16X4_F32.
//
// Roofline: ~27 GFLOP / ~40 MB => ~670 flop/byte, compute bound -> WMMA.
// fp32 WMMA chosen (not bf16/f16) because the expert gate is a hard
// threshold (acts0 > 0) that low-precision matmul error would flip.

typedef __attribute__((ext_vector_type(2))) float v2f;
typedef __attribute__((ext_vector_type(4))) float v4f;
typedef __attribute__((ext_vector_type(8))) float v8f;

namespace {
constexpr int kBatch = 2048;
constexpr int kDdata = 768;
constexpr int kDict0 = 128;   // == N_SAE
constexpr int kDict  = 32;
constexpr int kJK    = 128 * 32;  // 4096 flattened expert features

// GEMM tiling: 256 threads = 8 waves, 128x64 C tile, K staged 16 wide.
constexpr int BM = 128;
constexpr int BN = 64;
constexpr int TK = 16;
constexpr int LDS_K = TK + 2;  // 18: even stride -> 8B-aligned v2f frag reads,
                               // conflict-free across both 16-lane groups.
}

// One K-segment of C[m0:m0+128, n0:n0+64] += A[:,K] * B[K,:].
// EXPERT_B: B is W_enc1-style column-blocked: (d, n) at (n>>5)*768*32 + d*32 + (n&31)
// GATE_A:   A values are binarized (v>0 ? 1 : 0) while staging (gate @ b_dec1 term)
template<bool EXPERT_B, bool GATE_A>
__device__ __forceinline__ void gemm_seg(
    const float* __restrict__ A, int lda,
    const float* __restrict__ Bg, int ldb,
    int K, int m0, int n0,
    float (&As)[BM][LDS_K], float (&Bs)[BN][LDS_K],
    v8f (&acc)[2][2])
{
  const int tid  = threadIdx.x;
  const int lane = tid & 31;
  const int wave = tid >> 5;
  const int wm  = (wave & 3) * 32;      // wave's M offset in tile (4 waves down)
  const int wn  = (wave >> 2) * 32;     // wave's N offset in tile (2 waves across)
  const int l15 = lane & 15;
  const int kb2 = (lane >> 4) << 1;     // lanes 0-15 -> K+0/1, lanes 16-31 -> K+2/3

  const int arow = tid >> 1;            // 0..127
  const int acol = (tid & 1) * 8;       // 0 or 8
  const int brow = tid >> 4;            // 0..15
  const int bcol = (tid & 15) * 4;      // 0..60

  for (int k0 = 0; k0 < K; k0 += TK) {
    // ---- stage A tile (BM x TK), fragment-major As[m][k] ----
    {
      const float* ap = A + (size_t)(m0 + arow) * lda + (k0 + acol);
      v4f a0 = *(const v4f*)ap;
      v4f a1 = *(const v4f*)(ap + 4);
      if (GATE_A) {
#pragma unroll
        for (int i = 0; i < 4; ++i) {
          a0[i] = a0[i] > 0.f ? 1.f : 0.f;
          a1[i] = a1[i] > 0.f ? 1.f : 0.f;
        }
      }
#pragma unroll
      for (int i = 0; i < 4; ++i) {
        As[arow][acol + i]     = a0[i];
        As[arow][acol + 4 + i] = a1[i];
      }
    }
    // ---- stage B tile (TK x BN), transposed into Bs[n][k] ----
    {
      const int n = n0 + bcol;
      size_t off;
      if (EXPERT_B)
        off = (size_t)(n >> 5) * (kDdata * kDict) + (size_t)(k0 + brow) * kDict + (n & 31);
      else
        off = (size_t)(k0 + brow) * ldb + n;
      v4f bv = *(const v4f*)(Bg + off);
#pragma unroll
      for (int i = 0; i < 4; ++i) Bs[bcol + i][brow] = bv[i];
    }
    __syncthreads();

#pragma unroll
    for (int kk = 0; kk < TK; kk += 4) {
      // A frag layout (16x4 f32): VGPR0 = {lanes0-15:K+0 | lanes16-31:K+2},
      // VGPR1 = {K+1 | K+3}; M = lane&15. B mirrored with N = lane&15.
      v2f af[2], bf[2];
#pragma unroll
      for (int t = 0; t < 2; ++t) {
        af[t] = *(const v2f*)&As[wm + t * 16 + l15][kk + kb2];
        bf[t] = *(const v2f*)&Bs[wn + t * 16 + l15][kk + kb2];
      }
#pragma unroll
      for (int mt = 0; mt < 2; ++mt)
#pragma unroll
        for (int nt = 0; nt < 2; ++nt)
          acc[mt][nt] = __builtin_amdgcn_wmma_f32_16x16x4_f32(
              false, af[mt], false, bf[nt], (short)0, acc[mt][nt], false, false);
    }
    __syncthreads();
  }
}

// c0[n] = sum_d b_dec0[d]*W_enc0[d,n];  c1[j*32+k] = sum_d b_dec1[j,d]*W_enc1[j,d,k]
__global__ __launch_bounds__(256) void hsae_precompute(
    const float* __restrict__ W_enc0, const float* __restrict__ b_dec0,
    const float* __restrict__ W_enc1, const float* __restrict__ b_dec1,
    float* __restrict__ c0, float* __restrict__ c1)
{
  int idx = blockIdx.x * 256 + threadIdx.x;
  if (idx < kDict0) {
    float s = 0.f;
    for (int d = 0; d < kDdata; ++d) s += b_dec0[d] * W_enc0[d * kDict0 + idx];
    c0[idx] = s;
  } else if (idx < kDict0 + kJK) {
    int n = idx - kDict0;
    int j = n >> 5, k = n & 31;
    const float* w  = W_enc1 + (size_t)j * kDdata * kDict + k;
    const float* bd = b_dec1 + (size_t)j * kDdata;
    float s = 0.f;
    for (int d = 0; d < kDdata; ++d) s += bd[d] * w[d * kDict];
    c1[n] = s;
  }
}

// acts0 = relu(x @ W_enc0 - c0 + b_enc0)    [2048 x 128]
__global__ __launch_bounds__(256) void hsae_enc0(
    const float* __restrict__ x, const float* __restrict__ W_enc0,
    const float* __restrict__ b_enc0, const float* __restrict__ c0,
    float* __restrict__ acts0)
{
  __shared__ float As[BM][LDS_K];
  __shared__ float Bs[BN][LDS_K];
  v8f acc[2][2] = {};
  const int m0 = blockIdx.y * BM;
  const int n0 = blockIdx.x * BN;
  gemm_seg<false, false>(x, kDdata, W_enc0, kDict0, kDdata, m0, n0, As, Bs, acc);

  const int lane = threadIdx.x & 31, wave = threadIdx.x >> 5;
  const int wm = (wave & 3) * 32, wn = (wave >> 2) * 32;
  const int l15 = lane & 15, mh = (lane >> 4) * 8;
#pragma unroll
  for (int mt = 0; mt < 2; ++mt)
#pragma unroll
    for (int nt = 0; nt < 2; ++nt) {
      const int n = n0 + wn + nt * 16 + l15;
      const float bias = b_enc0[n] - c0[n];
#pragma unroll
      for (int r = 0; r < 8; ++r) {
        const int m = m0 + wm + mt * 16 + mh + r;
        float v = acc[mt][nt][r] + bias;
        acts0[(size_t)m * kDict0 + n] = v > 0.f ? v : 0.f;
      }
    }
}

// acts1 = gate * relu(x @ W_enc1_flat - c1 + b_enc1)   [2048 x 4096]
__global__ __launch_bounds__(256) void hsae_enc1(
    const float* __restrict__ x, const float* __restrict__ W_enc1,
    const float* __restrict__ b_enc1, const float* __restrict__ c1,
    const float* __restrict__ acts0, float* __restrict__ acts1)
{
  __shared__ float As[BM][LDS_K];
  __shared__ float Bs[BN][LDS_K];
  v8f acc[2][2] = {};
  const int m0 = blockIdx.y * BM;
  const int n0 = blockIdx.x * BN;
  gemm_seg<true, false>(x, kDdata, W_enc1, 0, kDdata, m0, n0, As, Bs, acc);

  const int lane = threadIdx.x & 31, wave = threadIdx.x >> 5;
  const int wm = (wave & 3) * 32, wn = (wave >> 2) * 32;
  const int l15 = lane & 15, mh = (lane >> 4) * 8;
#pragma unroll
  for (int mt = 0; mt < 2; ++mt)
#pragma unroll
    for (int nt = 0; nt < 2; ++nt) {
      const int n = n0 + wn + nt * 16 + l15;
      const float bias = b_enc1[n] - c1[n];
      const int j = n >> 5;
#pragma unroll
      for (int r = 0; r < 8; ++r) {
        const int m = m0 + wm + mt * 16 + mh + r;
        float v = acc[mt][nt][r] + bias;
        v = v > 0.f ? v : 0.f;
        const float g = acts0[(size_t)m * kDict0 + j];
        acts1[(size_t)m * kJK + n] = (g > 0.f) ? v : 0.f;
      }
    }
}

// out = acts1@W_dec1_flat + acts0@W_dec0 + gate(acts0)@b_dec1 + b_dec0  [2048 x 768]
__global__ __launch_bounds__(256) void hsae_dec(
    const float* __restrict__ acts1, const float* __restrict__ W_dec1,
    const float* __restrict__ acts0, const float* __restrict__ W_dec0,
    const float* __restrict__ b_dec1, const float* __restrict__ b_dec0,
    float* __restrict__ out)
{
  __shared__ float As[BM][LDS_K];
  __shared__ float Bs[BN][LDS_K];
  v8f acc[2][2] = {};
  const int m0 = blockIdx.y * BM;
  const int n0 = blockIdx.x * BN;
  // K segment 1: acts1 [2048,4096] x W_dec1 flat [4096,768]
  gemm_seg<false, false>(acts1, kJK, W_dec1, kDdata, kJK, m0, n0, As, Bs, acc);
  // K segment 2: acts0 [2048,128] x W_dec0 [128,768]
  gemm_seg<false, false>(acts0, kDict0, W_dec0, kDdata, kDict0, m0, n0, As, Bs, acc);
  // K segment 3: gate(acts0) x b_dec1 [128,768]
  gemm_seg<false, true >(acts0, kDict0, b_dec1, kDdata, kDict0, m0, n0, As, Bs, acc);

  const int lane = threadIdx.x & 31, wave = threadIdx.x >> 5;
  const int wm = (wave & 3) * 32, wn = (wave >> 2) * 32;
  const int l15 = lane & 15, mh = (lane >> 4) * 8;
#pragma unroll
  for (int mt = 0; mt < 2; ++mt)
#pragma unroll
    for (int nt = 0; nt < 2; ++nt) {
      const int n = n0 + wn + nt * 16 + l15;
      const float bias = b_dec0[n];
#pragma unroll
      for (int r = 0; r < 8; ++r) {
        const int m = m0 + wm + mt * 16 + mh + r;
        out[(size_t)m * kDdata + n] = acc[mt][nt][r] + bias;
      }
    }
}

extern "C" void kernel_launch(void* const* d_in, const int* /*in_sizes*/, int /*n_in*/,
                              void* d_out, int /*out_size*/, void* d_ws, size_t /*ws_size*/,
                              hipStream_t stream)
{
  const float* x      = (const float*)d_in[0];
  const float* W_enc0 = (const float*)d_in[1];
  const float* b_enc0 = (const float*)d_in[2];
  const float* W_dec0 = (const float*)d_in[3];
  const float* b_dec0 = (const float*)d_in[4];
  const float* W_enc1 = (const float*)d_in[5];
  const float* b_enc1 = (const float*)d_in[6];
  const float* W_dec1 = (const float*)d_in[7];
  const float* b_dec1 = (const float*)d_in[8];
  float* out = (float*)d_out;

  // workspace layout (floats): c0[128] | c1[4096] | pad | acts0[2048*128] | acts1[2048*4096]
  float* wsf   = (float*)d_ws;
  float* c0    = wsf;
  float* c1    = wsf + 128;
  float* acts0 = wsf + 4352;
  float* acts1 = wsf + 4352 + (size_t)kBatch * kDict0;   // ~34.6 MB total

  hsae_precompute<<<dim3((kDict0 + kJK + 255) / 256), dim3(256), 0, stream>>>(
      W_enc0, b_dec0, W_enc1, b_dec1, c0, c1);
  hsae_enc0<<<dim3(kDict0 / BN, kBatch / BM), dim3(256), 0, stream>>>(
      x, W_enc0, b_enc0, c0, acts0);
  hsae_enc1<<<dim3(kJK / BN, kBatch / BM), dim3(256), 0, stream>>>(
      x, W_enc1, b_enc1, c1, acts0, acts1);
  hsae_dec<<<dim3(kDdata / BN, kBatch / BM), dim3(256), 0, stream>>>(
      acts1, W_dec1, acts0, W_dec0, b_dec1, b_dec0, out);
}